// GC_GNN_59356448031329
// MI455X (gfx1250) — compile-verified
//
#include <hip/hip_runtime.h>
#include <hip/hip_bf16.h>

#define NN 100000
#define NE 1600000
#define HD 64
#define NG 128
#define NL 7

typedef __attribute__((ext_vector_type(2))) float v2f;
typedef __attribute__((ext_vector_type(8))) float v8f;

// ---------- native CDNA5 float atomics (no-return, STOREcnt-tracked) ----------
__device__ __forceinline__ void atomic_max_f32(float* p, float v) {
    asm volatile("global_atomic_max_num_f32 %0, %1, off scope:SCOPE_DEV"
                 :: "v"(p), "v"(v) : "memory");
}
__device__ __forceinline__ void atomic_add_f32(float* p, float v) {
    asm volatile("global_atomic_add_f32 %0, %1, off scope:SCOPE_DEV"
                 :: "v"(p), "v"(v) : "memory");
}

// ---------- async global -> LDS copy (ASYNCcnt-tracked), 16B per lane ----------
// LDS operand: generic-pointer low 32 bits == wave-relative LDS byte offset
// (ISA 10.2 aperture rule: the low 32 address bits give the LDS address).
__device__ __forceinline__ void async_copy_b128(const float* g, const float* lds) {
    unsigned l = (unsigned)(uintptr_t)lds;
    asm volatile("global_load_async_to_lds_b128 %0, %1, off"
                 :: "v"(l), "v"(g) : "memory");
}
__device__ __forceinline__ void wait_asynccnt0() {
    asm volatile("s_wait_asynccnt 0x0" ::: "memory");
}

// ---------- fill agg with -inf ----------
__global__ void __launch_bounds__(256) k_fill_neginf(float4* p, int n4) {
    int g = blockIdx.x * 256 + threadIdx.x;
    if (g < n4) {
        float ninf = -__builtin_inff();
        p[g] = make_float4(ninf, ninf, ninf, ninf);
    }
}

__global__ void __launch_bounds__(256) k_zero(float* p, int n) {
    int g = blockIdx.x * 256 + threadIdx.x;
    if (g < n) p[g] = 0.0f;
}

// ---------- edge-parallel scatter-max: agg[dst] = max(agg[dst], h[src]*w) ----------
// 16 threads per edge, float4 per thread -> 256B coalesced gather per edge row.
__global__ void __launch_bounds__(256) k_scatter_max(
    const float* __restrict__ h, const int* __restrict__ src,
    const int* __restrict__ dst, const float* __restrict__ ew,
    float* __restrict__ agg, int nEdges)
{
    int g = blockIdx.x * 256 + threadIdx.x;
    int e = g >> 4;
    if (e >= nEdges) return;
    int dc = (g & 15) * 4;
    float w = ew[e];
    const float4 v = *(const float4*)(h + (size_t)src[e] * HD + dc);
    float* ap = agg + (size_t)dst[e] * HD + dc;
    atomic_max_f32(ap + 0, v.x * w);
    atomic_max_f32(ap + 1, v.y * w);
    atomic_max_f32(ap + 2, v.z * w);
    atomic_max_f32(ap + 3, v.w * w);
}

// ---------- fused WMMA layer GEMM: hout = act(max0(agg)@Wrel + hin@Wroot + b) ----------
// One wave computes a 16x64 output tile. 8 waves / block -> 128 rows / block.
// Weights staged via async global->LDS (straight layout), then repacked in
// place to K-pair-interleaved: sW[m*4096 + (k/2)*128 + 2n + (k&1)], so each
// WMMA B fragment {W[ka][n], W[ka+1][n]} is one aligned ds_load_b64.
__global__ void __launch_bounds__(256) k_gconv_gemm(
    const float* __restrict__ agg, const float* __restrict__ hin,
    const float* __restrict__ Wrel, const float* __restrict__ Wroot,
    const float* __restrict__ bias, float* __restrict__ hout, int relu)
{
    __shared__ float sW[2 * HD * HD];     // 32 KB: [0,4096)=Wrel, [4096,8192)=Wroot

    // 1) async copy, straight row-major layout (coalesced 16B per lane)
    for (int c = threadIdx.x; c < 2048; c += 256) {
        const float* g = (c < 1024) ? (Wrel + c * 4) : (Wroot + (c - 1024) * 4);
        async_copy_b128(g, &sW[c * 4]);
    }
    wait_asynccnt0();
    __syncthreads();

    // 2) in-place repack to pair-interleaved via registers
    float tmp[32];
    #pragma unroll
    for (int c = 0; c < 32; ++c) tmp[c] = sW[threadIdx.x + c * 256];
    __syncthreads();
    #pragma unroll
    for (int c = 0; c < 32; ++c) {
        int i = threadIdx.x + c * 256;
        int m = i >> 12, k = (i >> 6) & 63, n = i & 63;
        sW[m * 4096 + (k >> 1) * 128 + (n << 1) + (k & 1)] = tmp[c];
    }
    __syncthreads();

    const int lane = threadIdx.x & 31;
    const int wave = threadIdx.x >> 5;
    const int m0 = (blockIdx.x * 8 + wave) * 16;
    if (m0 >= NN) return;                 // wave-uniform: EXEC stays all-ones

    const int r  = lane & 15;             // M row within tile / N col within n-tile
    const int kh = lane >> 4;             // K-half select (ISA 16x4 A layout)
    const float* aggRow = agg + (size_t)(m0 + r) * HD;
    const float* hRow   = hin + (size_t)(m0 + r) * HD;
    const float ninf = -__builtin_inff();

    v8f c[4] = {v8f{}, v8f{}, v8f{}, v8f{}};

    #pragma unroll
    for (int kb = 0; kb < HD; kb += 4) {
        const int ka = kb + 2 * kh;       // lane's K pair start
        const int p  = (kb >> 1) + kh;    // K-pair row in packed LDS
        float a0 = aggRow[ka], a1 = aggRow[ka + 1];
        a0 = (a0 == ninf) ? 0.0f : a0;    // empty-segment fill (PyG semantics)
        a1 = (a1 == ninf) ? 0.0f : a1;
        v2f aA; aA.x = a0;       aA.y = a1;
        v2f aH; aH.x = hRow[ka]; aH.y = hRow[ka + 1];

        const float* wbase = &sW[p * 128 + (r << 1)];
        #pragma unroll
        for (int j = 0; j < 4; ++j) {
            v2f bR = *(const v2f*)(wbase + j * 32);          // Wrel pair
            v2f bT = *(const v2f*)(wbase + 4096 + j * 32);   // Wroot pair
            c[j] = __builtin_amdgcn_wmma_f32_16x16x4_f32(
                       false, aA, false, bR, (short)0, c[j], false, false);
            c[j] = __builtin_amdgcn_wmma_f32_16x16x4_f32(
                       false, aH, false, bT, (short)0, c[j], false, false);
        }
    }

    #pragma unroll
    for (int j = 0; j < 4; ++j) {
        const int col = j * 16 + r;
        const float bj = bias[col];
        #pragma unroll
        for (int vv = 0; vv < 8; ++vv) {  // C/D layout: M = vv + 8*kh, N = col
            float val = c[j][vv] + bj;
            if (relu) val = fmaxf(val, 0.0f);
            hout[(size_t)(m0 + vv + 8 * kh) * HD + col] = val;
        }
    }
}

// ---------- pooled[batch[n]] += h[n] ----------
__global__ void __launch_bounds__(256) k_pool_sum(
    const float* __restrict__ h, const int* __restrict__ batch,
    float* __restrict__ pooled, int nNodes)
{
    int g = blockIdx.x * 256 + threadIdx.x;
    int n = g >> 4;
    if (n >= nNodes) return;
    int dc = (g & 15) * 4;
    const float4 v = *(const float4*)(h + (size_t)n * HD + dc);
    float* pp = pooled + (size_t)batch[n] * HD + dc;
    atomic_add_f32(pp + 0, v.x);
    atomic_add_f32(pp + 1, v.y);
    atomic_add_f32(pp + 2, v.z);
    atomic_add_f32(pp + 3, v.w);
}

// ---------- out[128x2] = pooled @ linW + linb ----------
__global__ void __launch_bounds__(256) k_final_linear(
    const float* __restrict__ pooled, const float* __restrict__ linW,
    const float* __restrict__ linb, float* __restrict__ out)
{
    int t = threadIdx.x;                  // 256 = 128 graphs x 2 classes
    int gph = t >> 1, cls = t & 1;
    float acc = linb[cls];
    #pragma unroll
    for (int k = 0; k < HD; ++k)
        acc += pooled[gph * HD + k] * linW[k * 2 + cls];
    out[gph * 2 + cls] = acc;
}

extern "C" void kernel_launch(void* const* d_in, const int* in_sizes, int n_in,
                              void* d_out, int out_size, void* d_ws, size_t ws_size,
                              hipStream_t stream) {
    const float* x     = (const float*)d_in[0];
    const int*   ei    = (const int*)  d_in[1];
    const int*   batch = (const int*)  d_in[2];
    const float* ew    = (const float*)d_in[3];
    const float* Wrel  = (const float*)d_in[4];
    const float* Wroot = (const float*)d_in[5];
    const float* bias  = (const float*)d_in[6];
    const float* linW  = (const float*)d_in[7];
    const float* linb  = (const float*)d_in[8];
    float* out = (float*)d_out;

    const int* src = ei;
    const int* dst = ei + NE;

    float* hA     = (float*)d_ws;                  // NN*HD
    float* hB     = hA  + (size_t)NN * HD;         // NN*HD
    float* agg    = hB  + (size_t)NN * HD;         // NN*HD
    float* pooled = agg + (size_t)NN * HD;         // NG*HD

    const int gridFill    = (NN * HD / 4 + 255) / 256;   // 6250
    const int gridScatter = (NE * 16 + 255) / 256;       // 100000
    const int gridGemm    = (NN / 16 + 7) / 8;           // 782
    const int gridPool    = (NN * 16 + 255) / 256;       // 6250

    const float* hin = x;
    float* hout = hA;
    for (int l = 0; l < NL; ++l) {
        k_fill_neginf<<<gridFill, 256, 0, stream>>>((float4*)agg, NN * HD / 4);
        k_scatter_max<<<gridScatter, 256, 0, stream>>>(hin, src, dst, ew, agg, NE);
        k_gconv_gemm<<<gridGemm, 256, 0, stream>>>(
            agg, hin, Wrel + (size_t)l * HD * HD, Wroot + (size_t)l * HD * HD,
            bias + (size_t)l * HD, hout, (l < NL - 1) ? 1 : 0);
        hin = hout;
        hout = (hout == hA) ? hB : hA;
    }

    k_zero<<<(NG * HD + 255) / 256, 256, 0, stream>>>(pooled, NG * HD);
    k_pool_sum<<<gridPool, 256, 0, stream>>>(hin, batch, pooled, NN);
    k_final_linear<<<1, 256, 0, stream>>>(pooled, linW, linb, out);
}